// TriPlaneEncoder_49649821942507
// MI455X (gfx1250) — compile-verified
//
#include <hip/hip_runtime.h>
#include <hip/hip_bf16.h>

typedef __attribute__((ext_vector_type(2))) float v2f;
typedef __attribute__((ext_vector_type(4))) float v4f;
typedef __attribute__((ext_vector_type(8))) float v8f;

#define RES    512
#define NPIX   (RES * RES)          // 262144
#define NFEAT  16                   // channels per plane
#define KTOT   48                   // 3 * NFEAT
#define OUTC   32
#define LDS_STRIDE 52               // 48 floats padded (odd multiple of 4) to spread banks

// ---------------------------------------------------------------------------
// One-time layout transform: [C,H,W] -> [H,W,C] so one texel's 16 channels are
// a contiguous 64B chunk (4 x b128 loads per bilinear corner).
// Reads coalesced across lanes; writes are 64B/thread contiguous chunks.
// ---------------------------------------------------------------------------
__global__ __launch_bounds__(256) void transpose_planes_kernel(
    const float* __restrict__ xy, const float* __restrict__ yz,
    const float* __restrict__ xz, float* __restrict__ ws) {
  int tid   = blockIdx.x * blockDim.x + threadIdx.x;   // 0 .. 3*NPIX-1
  int plane = tid >> 18;                               // NPIX == 2^18
  int pix   = tid & (NPIX - 1);
  const float* src = (plane == 0) ? xy : ((plane == 1) ? yz : xz);
  float* dst = ws + (size_t)plane * ((size_t)NPIX * NFEAT) + (size_t)pix * NFEAT;
#pragma unroll
  for (int q = 0; q < 4; ++q) {
    v4f v;
    v.x = src[(size_t)(4 * q + 0) * NPIX + pix];
    v.y = src[(size_t)(4 * q + 1) * NPIX + pix];
    v.z = src[(size_t)(4 * q + 2) * NPIX + pix];
    v.w = src[(size_t)(4 * q + 3) * NPIX + pix];
    *(v4f*)(dst + 4 * q) = v;
  }
}

// ---------------------------------------------------------------------------
// Bilinear gather helpers
// ---------------------------------------------------------------------------
template <bool TRANSPOSED>
__device__ __forceinline__ void accum_corner(const float* __restrict__ pl,
                                             int pix, float wgt, float* f) {
  if constexpr (TRANSPOSED) {
    const v4f* t = (const v4f*)(pl + (size_t)pix * NFEAT);
#pragma unroll
    for (int q = 0; q < 4; ++q) {
      v4f v = t[q];
      f[q * 4 + 0] = fmaf(wgt, v.x, f[q * 4 + 0]);
      f[q * 4 + 1] = fmaf(wgt, v.y, f[q * 4 + 1]);
      f[q * 4 + 2] = fmaf(wgt, v.z, f[q * 4 + 2]);
      f[q * 4 + 3] = fmaf(wgt, v.w, f[q * 4 + 3]);
    }
  } else {
#pragma unroll
    for (int c = 0; c < NFEAT; ++c)
      f[c] = fmaf(wgt, pl[(size_t)c * NPIX + pix], f[c]);
  }
}

template <bool TRANSPOSED>
__device__ __forceinline__ void sample_plane(const float* __restrict__ pl,
                                             float u, float v, float* f) {
  // align_corners=False: pix = ((g+1)*512 - 1)/2 = 256*g + 255.5
  float ix = fmaf(u, 256.0f, 255.5f);
  float iy = fmaf(v, 256.0f, 255.5f);
  float x0f = floorf(ix), y0f = floorf(iy);
  float wx1 = ix - x0f,  wy1 = iy - y0f;
  float wx0 = 1.0f - wx1, wy0 = 1.0f - wy1;
  int x0 = (int)x0f, y0 = (int)y0f;
  int x1 = x0 + 1,   y1 = y0 + 1;
  // padding_mode='zeros': out-of-range corners contribute 0
  float vx0 = (x0 >= 0 && x0 < RES) ? 1.0f : 0.0f;
  float vx1 = (x1 >= 0 && x1 < RES) ? 1.0f : 0.0f;
  float vy0 = (y0 >= 0 && y0 < RES) ? 1.0f : 0.0f;
  float vy1 = (y1 >= 0 && y1 < RES) ? 1.0f : 0.0f;
  int cx0 = min(max(x0, 0), RES - 1), cx1 = min(max(x1, 0), RES - 1);
  int cy0 = min(max(y0, 0), RES - 1), cy1 = min(max(y1, 0), RES - 1);
#pragma unroll
  for (int c = 0; c < NFEAT; ++c) f[c] = 0.0f;
  accum_corner<TRANSPOSED>(pl, cy0 * RES + cx0, wx0 * wy0 * vx0 * vy0, f);
  accum_corner<TRANSPOSED>(pl, cy0 * RES + cx1, wx1 * wy0 * vx1 * vy0, f);
  accum_corner<TRANSPOSED>(pl, cy1 * RES + cx0, wx0 * wy1 * vx0 * vy1, f);
  accum_corner<TRANSPOSED>(pl, cy1 * RES + cx1, wx1 * wy1 * vx1 * vy1, f);
}

// ---------------------------------------------------------------------------
// Main kernel: 256 threads = 8 waves; each wave owns 32 points.
// Per wave: gather 32x48 feats -> LDS -> two 16x4-f32 A tiles per k-step,
// 12 k-steps x 2 output-column tiles of V_WMMA_F32_16X16X4_F32.
// ---------------------------------------------------------------------------
template <bool TRANSPOSED>
__global__ __launch_bounds__(256) void triplane_wmma_kernel(
    const float* __restrict__ x,    // [N,3]
    const float* __restrict__ p0,   // xy plane base
    const float* __restrict__ p1,   // yz plane base
    const float* __restrict__ p2,   // xz plane base
    const float* __restrict__ w,    // [48,32] row-major
    float* __restrict__ out,        // [N,32]
    int n) {
  __shared__ float lds[8 * 32 * LDS_STRIDE];

  const int lane = threadIdx.x & 31;
  const int wave = threadIdx.x >> 5;
  const int base = (blockIdx.x * 8 + wave) * 32;   // first point of this wave
  if (blockIdx.x * 256 >= n) return;               // uniform, full-block guard
  const int point = base + lane;

  // ---- per-lane point + tri-plane bilinear gather (48 features) ----
  float gx = x[(size_t)point * 3 + 0];
  float gy = x[(size_t)point * 3 + 1];
  float gz = x[(size_t)point * 3 + 2];

  float feats[KTOT];
  sample_plane<TRANSPOSED>(p0, gx, gy, feats + 0);    // xy
  sample_plane<TRANSPOSED>(p1, gy, gz, feats + 16);   // yz
  sample_plane<TRANSPOSED>(p2, gx, gz, feats + 32);   // xz

  // ---- stage feats to LDS (row = point-in-tile, padded stride) ----
  float* myrow = lds + (size_t)(wave * 32 + lane) * LDS_STRIDE;
#pragma unroll
  for (int q = 0; q < 12; ++q) {
    v4f v; v.x = feats[4*q+0]; v.y = feats[4*q+1];
           v.z = feats[4*q+2]; v.w = feats[4*q+3];
    *(v4f*)(myrow + 4 * q) = v;
  }

  // ---- preload B fragments (48x32 weight, ISA 4x16 f32 B layout):
  //      VGPR0: lanes0-15 K=k0 / lanes16-31 K=k0+2 ; VGPR1: K=k0+1 / K=k0+3
  const int off = (lane < 16) ? 0 : 2;
  const int col = lane & 15;
  v2f Bf[12][2];
#pragma unroll
  for (int kb = 0; kb < 12; ++kb) {
#pragma unroll
    for (int ct = 0; ct < 2; ++ct) {
      int c = ct * 16 + col;
      Bf[kb][ct].x = w[(size_t)(4 * kb + off)     * OUTC + c];
      Bf[kb][ct].y = w[(size_t)(4 * kb + off + 1) * OUTC + c];
    }
  }

  __syncthreads();   // LDS feats visible / ordered before A-fragment reads

  // ---- WMMA accumulation: D[32x32] = feats[32x48] * W[48x32] ----
  v8f acc[2][2] = {};
  const float* wbase = lds + (size_t)wave * 32 * LDS_STRIDE;
  const int m = lane & 15;   // A-matrix M index (both half-waves)
#pragma unroll
  for (int kb = 0; kb < 12; ++kb) {
#pragma unroll
    for (int rt = 0; rt < 2; ++rt) {
      // A 16x4 f32 layout: VGPR0 = K=(off), VGPR1 = K=(off+1) -> one b64 read
      v2f A = *(const v2f*)(wbase + (size_t)(rt * 16 + m) * LDS_STRIDE +
                            4 * kb + off);
      acc[rt][0] = __builtin_amdgcn_wmma_f32_16x16x4_f32(
          false, A, false, Bf[kb][0], (short)0, acc[rt][0], false, false);
      acc[rt][1] = __builtin_amdgcn_wmma_f32_16x16x4_f32(
          false, A, false, Bf[kb][1], (short)0, acc[rt][1], false, false);
    }
  }

  // ---- store D tiles: VGPR v, lanes0-15 -> M=v, lanes16-31 -> M=v+8 ----
  const int rowoff = (lane < 16) ? 0 : 8;
#pragma unroll
  for (int rt = 0; rt < 2; ++rt) {
#pragma unroll
    for (int ct = 0; ct < 2; ++ct) {
#pragma unroll
      for (int v = 0; v < 8; ++v) {
        int pr = base + rt * 16 + rowoff + v;
        out[(size_t)pr * OUTC + ct * 16 + col] = acc[rt][ct][v];
      }
    }
  }
}

// ---------------------------------------------------------------------------
// Launch
// ---------------------------------------------------------------------------
extern "C" void kernel_launch(void* const* d_in, const int* in_sizes, int n_in,
                              void* d_out, int out_size, void* d_ws, size_t ws_size,
                              hipStream_t stream) {
  const float* x  = (const float*)d_in[0];   // [1, N, 3]
  const float* xy = (const float*)d_in[1];   // [1, 16, 512, 512]
  const float* yz = (const float*)d_in[2];
  const float* xz = (const float*)d_in[3];
  const float* w  = (const float*)d_in[4];   // [1, 48, 32]
  float* out = (float*)d_out;                // [1, N, 32]

  const int n = in_sizes[0] / 3;             // number of points
  const size_t planeElems = (size_t)NPIX * NFEAT;
  const size_t needWs = 3 * planeElems * sizeof(float);

  const int mainBlocks = n / 256;            // 256 points per block (8 waves x 32)

  if (ws_size >= needWs) {
    float* wsT = (float*)d_ws;
    transpose_planes_kernel<<<(3 * NPIX) / 256, 256, 0, stream>>>(xy, yz, xz, wsT);
    triplane_wmma_kernel<true><<<mainBlocks, 256, 0, stream>>>(
        x, wsT, wsT + planeElems, wsT + 2 * planeElems, w, out, n);
  } else {
    triplane_wmma_kernel<false><<<mainBlocks, 256, 0, stream>>>(
        x, xy, yz, xz, w, out, n);
  }
}